// FFM_Layer_7215545057762
// MI455X (gfx1250) — compile-verified
//
#include <hip/hip_runtime.h>

// CDNA5 / gfx1250, wave32 only.
typedef __attribute__((ext_vector_type(2))) float v2f;
typedef __attribute__((ext_vector_type(8))) float v8f;

#define FIELD_NUM 20
#define VOCAB     50000
#define KDIM      8
#define ROWLEN    (FIELD_NUM * KDIM)   // 160 f32 per gathered v-row

// One wave (32 lanes) per batch element.
// lv[160] = column sums of the gathered 20x160 matrix, computed with
// V_WMMA_F32_16X16X4_F32: A = ones(16x4), B = 4x16 chunk, C accumulates.
// Exact f32 adds -> numerically identical to the reference reduction.
__global__ __launch_bounds__(32)
void ffm_layer_kernel(const int* __restrict__ inputs,
                      const float* __restrict__ w0,
                      const float* __restrict__ w,
                      const float* __restrict__ v,
                      float* __restrict__ out)
{
    const int b     = blockIdx.x;
    const int lane  = threadIdx.x;   // 0..31
    const int laneN = lane & 15;     // column-in-tile
    const bool hi   = lane >= 16;

    // ---- wave-uniform part: index math, w-gathers (scalar loads) ----
    const int* inRow = inputs + b * FIELD_NUM;
    float first = w0[0];
    const float* rowp[FIELD_NUM];
#pragma unroll
    for (int f = 0; f < FIELD_NUM; ++f) {
        const int idx = inRow[f] + f * VOCAB;          // global feature id
        first += w[idx];                               // first-order gather
        rowp[f] = v + (size_t)idx * ROWLEN;            // base of 160-f32 row
    }

    const v2f ones = {1.0f, 1.0f};                     // A(16x4) all-ones

    float p   = 0.0f;   // per-lane: sum of lv over this lane's column slots
    float ssq = 0.0f;   // per-lane: sum of lv^2 (wave total = 2 * sum lv^2)

#pragma unroll
    for (int ct = 0; ct < 10; ++ct) {                  // 10 tiles of 16 cols
        const int cbase = ct * 16 + laneN;
        v8f acc = {};
#pragma unroll
        for (int kc = 0; kc < 5; ++kc) {               // 5 chunks of 4 rows
            // B fragment 4x16 f32:
            //   VGPR0: lanes 0-15 = row 4kc+0, lanes 16-31 = row 4kc+2
            //   VGPR1: lanes 0-15 = row 4kc+1, lanes 16-31 = row 4kc+3
            const float* r0 = hi ? rowp[4 * kc + 2] : rowp[4 * kc + 0];
            const float* r1 = hi ? rowp[4 * kc + 3] : rowp[4 * kc + 1];
            v2f bfrag;
            bfrag.x = r0[cbase];                       // 64B coalesced / half-wave
            bfrag.y = r1[cbase];
            acc = __builtin_amdgcn_wmma_f32_16x16x4_f32(
                /*neg_a=*/false, ones,
                /*neg_b=*/false, bfrag,
                /*c_mod=*/(short)0, acc,
                /*reuse_a=*/false, /*reuse_b=*/false);
        }
        // D: every M-row identical; lane n VGPR0 = lv[ct*16 + (n&15)],
        // so each column value appears exactly twice across the wave.
        const float lv = acc[0];
        p   += lv;
        ssq += lv * lv;
    }

    // All columns owned by lane n satisfy col%8 == n%8, so
    // t = p + p(lane^8) equals s[k=lane%8]; each s[k] appears 4x in the wave.
    float t  = p + __shfl_xor(p, 8, 32);
    float s2 = t * t;

    // Full-wave butterfly sums.
#pragma unroll
    for (int m = 16; m >= 1; m >>= 1) {
        s2  += __shfl_xor(s2,  m, 32);   // wave sum = 4 * sum_k s[k]^2
        ssq += __shfl_xor(ssq, m, 32);   // wave sum = 2 * sum lv^2
    }

    if (lane == 0) {
        const float second = 0.125f * s2 - 0.25f * ssq;  // 0.5*(S1/4 - S2/2)
        out[b] = first + second;
    }
}

extern "C" void kernel_launch(void* const* d_in, const int* in_sizes, int n_in,
                              void* d_out, int out_size, void* d_ws, size_t ws_size,
                              hipStream_t stream)
{
    const int*   inputs = (const int*)  d_in[0];  // [4096, 20] int32
    const float* w0     = (const float*)d_in[1];  // [1]
    const float* w      = (const float*)d_in[2];  // [1,000,000, 1]
    const float* v      = (const float*)d_in[3];  // [1,000,000, 20, 8]
    float* out = (float*)d_out;                   // [4096, 1]

    ffm_layer_kernel<<<4096, 32, 0, stream>>>(inputs, w0, w, v, out);
}